// GIN_13632226197772
// MI455X (gfx1250) — compile-verified
//
#include <hip/hip_runtime.h>
#include <cstdint>

// ---------------------------------------------------------------------------
// Problem constants (fixed by the reference)
// ---------------------------------------------------------------------------
namespace {
constexpr int NN      = 100000;  // nodes
constexpr int EE      = 800000;  // edges
constexpr int BB      = 100;     // graphs
constexpr int NSTRIDE = 100;     // feature-row stride in floats (400B, 16B aligned)
constexpr int MAXD    = 100;     // max feature dim
constexpr int TR      = 64;      // GEMM rows per block (4 waves x 16)
constexpr int WCOLS   = 112;     // padded weight columns (7*16)
constexpr int GNPAD   = 104;     // padded per-graph dim for LDS reductions

typedef __attribute__((ext_vector_type(2))) float v2f;  // WMMA f32 A/B frag
typedef __attribute__((ext_vector_type(8))) float v8f;  // WMMA f32 C/D frag
}

// ---------------------------------------------------------------------------
// Tensor Data Mover support (this toolchain: 6-arg builtin,
//   (u32x4 g0, i32x8 g1, i32x4 g2, i32x4 g3, i32x8 extra, i32 cpol));
// fallback = manual copy if the builtin is missing.
// ---------------------------------------------------------------------------
#if defined(__has_builtin)
#if __has_builtin(__builtin_amdgcn_tensor_load_to_lds) && \
    __has_builtin(__builtin_amdgcn_s_wait_tensorcnt)
#define HAVE_TDM 1
#endif
#endif
#ifndef HAVE_TDM
#define HAVE_TDM 0
#endif

#if HAVE_TDM
typedef __attribute__((ext_vector_type(4))) unsigned int u32x4;
typedef __attribute__((ext_vector_type(8))) int          i32x8;
typedef __attribute__((ext_vector_type(4))) int          i32x4;

// 2D tile load: tile_w x tile_h elements (f32), global row stride = stride_elems,
// OOB rows (>= tensor_h) return zero. LDS receives the tile compacted
// (row pitch = tile_w), i.e. pitch must equal tile_w at the consumer.
__device__ __forceinline__ void tdm_load_2d(unsigned lds_addr, const float* gptr,
                                            int tile_w, int tile_h,
                                            int tensor_w, int tensor_h,
                                            int stride_elems) {
  unsigned long long ga = (unsigned long long)(uintptr_t)gptr;
  u32x4 g0;
  g0[0] = 1u;                                        // count=1, user descriptor
  g0[1] = lds_addr;                                  // LDS byte address
  g0[2] = (unsigned)(ga & 0xFFFFFFFFull);            // global_addr[31:0]
  g0[3] = (unsigned)((ga >> 32) & 0x1FFFFFFull) | (2u << 30);  // [56:32] | type=2
  i32x8 g1;
  g1[0] = (int)(2u << 16);                           // data_size = 4 bytes
  g1[1] = (int)(((unsigned)tensor_w & 0xFFFFu) << 16);          // tensor_dim0 lo16
  g1[2] = (int)((((unsigned)tensor_w >> 16) & 0xFFFFu) |
                (((unsigned)tensor_h & 0xFFFFu) << 16));        // dim0 hi | dim1 lo
  g1[3] = (int)((((unsigned)tensor_h >> 16) & 0xFFFFu) |
                (((unsigned)tile_w & 0xFFFFu) << 16));          // dim1 hi | tile_dim0
  g1[4] = (int)((unsigned)tile_h & 0xFFFFu);                    // tile_dim1 (tile_dim2=0)
  g1[5] = (int)(unsigned)stride_elems;                          // tensor_dim0_stride lo32
  g1[6] = 0;                                                    // stride0 hi | stride1 lo
  g1[7] = 0;
  i32x4 z4 = {0, 0, 0, 0};                           // groups 2/3 unused (<=2D)
  i32x8 z8 = {0, 0, 0, 0, 0, 0, 0, 0};
  __builtin_amdgcn_tensor_load_to_lds(g0, g1, z4, z4, z8, 0);
}
#endif

// ---------------------------------------------------------------------------
// Small utility kernels
// ---------------------------------------------------------------------------
__global__ __launch_bounds__(256) void fill_kernel(float* __restrict__ p, float v, int n) {
  int i = blockIdx.x * blockDim.x + threadIdx.x;
  if (i < n) p[i] = v;
}

__global__ __launch_bounds__(256) void degree_kernel(const int* __restrict__ dst,
                                                     float* __restrict__ deg) {
  int e = blockIdx.x * blockDim.x + threadIdx.x;
  if (e < EE) atomicAdd(&deg[dst[e]], 1.0f);
}

__global__ __launch_bounds__(256) void gcount_kernel(const int* __restrict__ batch,
                                                     float* __restrict__ gcnt) {
  int n = blockIdx.x * blockDim.x + threadIdx.x;
  if (n < NN) atomicAdd(&gcnt[batch[n]], 1.0f);
}

__global__ __launch_bounds__(256) void init_h_kernel(const float* __restrict__ x,
                                                     float* __restrict__ h) {
  int n = blockIdx.x * blockDim.x + threadIdx.x;
  if (n < NN) h[n * NSTRIDE] = x[n];   // d_in = 1; only column 0 ever read
}

// Pack [Wl ; Wr] into a zero-padded, k-pair-interleaved panel:
//   Wp[(k>>1)*2*WCOLS + n*2 + (k&1)] = W'(k, n)
// so a B fragment (values (k,n),(k+1,n)) is one contiguous 8-byte load.
__global__ __launch_bounds__(256) void pack_w_kernel(
    const float* __restrict__ Wl, const float* __restrict__ Wr,
    float* __restrict__ Wp, int din, int dout, int KPP) {
  int i = blockIdx.x * blockDim.x + threadIdx.x;
  int total = 2 * KPP * WCOLS;
  if (i >= total) return;
  int k = i / WCOLS;
  int n = i - k * WCOLS;
  const float* W = (k < KPP) ? Wl : Wr;
  int kk = (k < KPP) ? k : (k - KPP);
  float v = (kk < din && n < dout) ? W[kk * dout + n] : 0.0f;
  Wp[(k >> 1) * (2 * WCOLS) + n * 2 + (k & 1)] = v;
}

// ---------------------------------------------------------------------------
// Edge aggregation: agg[dst] += h[src]   (mean applied later via deg)
// ---------------------------------------------------------------------------
__global__ __launch_bounds__(256) void aggregate_v4_kernel(
    const float* __restrict__ h, const int* __restrict__ src,
    const int* __restrict__ dst, float* __restrict__ agg) {
  int e = blockIdx.x * blockDim.x + threadIdx.x;
  int c = blockIdx.y;                       // float4 chunk index
  if (e >= EE) return;
  int s = src[e], d = dst[e];
  const float4 v = *(const float4*)&h[s * NSTRIDE + 4 * c];
  float* o = &agg[d * NSTRIDE + 4 * c];
  __builtin_prefetch(o, 0, 3);              // pull RMW line toward near cache
  atomicAdd(o + 0, v.x);
  atomicAdd(o + 1, v.y);
  atomicAdd(o + 2, v.z);
  atomicAdd(o + 3, v.w);
}

__global__ __launch_bounds__(256) void aggregate_s_kernel(
    const float* __restrict__ h, const int* __restrict__ src,
    const int* __restrict__ dst, float* __restrict__ agg) {
  int e = blockIdx.x * blockDim.x + threadIdx.x;
  int k = blockIdx.y;
  if (e >= EE) return;
  atomicAdd(&agg[dst[e] * NSTRIDE + k], h[src[e] * NSTRIDE + k]);
}

// ---------------------------------------------------------------------------
// Fused SAGE GEMM via full-precision WMMA:
//   hout = (agg * 1/deg) @ Wl + hin @ Wr + b
// CT (number of 16-wide output tiles) is a template parameter so the inner
// loop is branch-free: per k-step = 1 ds_load_b64 + CT global_load_b64 + CT
// wmma, fully unrolled -> the scheduler can batch loads ahead of the wmmas.
// A panels staged in LDS by the Tensor Data Mover when the tile is dense
// (din == KPP); degree scaling folded into the A fragment per lane.
// ---------------------------------------------------------------------------
template <int CT>
__global__ __launch_bounds__(128) void sage_gemm_wmma(
    const float* __restrict__ hin, const float* __restrict__ agg,
    const float* __restrict__ deg, const float* __restrict__ Wp,
    const float* __restrict__ bias, float* __restrict__ hout,
    int din, int dout, int KPP) {
  __shared__ float sAgg[TR * MAXD];   // 25.6 KB, pitch = KPP (conflict-free)
  __shared__ float sH[TR * MAXD];     // 25.6 KB
  const int row0 = blockIdx.x * TR;
  const int tid  = threadIdx.x;

#if HAVE_TDM
  if (din == KPP) {
    // Dense tile: one async TDM descriptor per panel, issued by wave 0.
    // tensor_dim1 = NN - row0 makes the DMA zero-fill rows past the last node.
    if (tid < 32) {
      tdm_load_2d((unsigned)(uintptr_t)sAgg, agg + (size_t)row0 * NSTRIDE,
                  din, TR, NSTRIDE, NN - row0, NSTRIDE);
      tdm_load_2d((unsigned)(uintptr_t)sH, hin + (size_t)row0 * NSTRIDE,
                  din, TR, NSTRIDE, NN - row0, NSTRIDE);
      __builtin_amdgcn_s_wait_tensorcnt(0);
    }
  } else
#endif
  {
    for (int idx = tid; idx < TR * KPP; idx += 128) {
      int r = idx / KPP;
      int k = idx - r * KPP;
      int node = row0 + r;
      float va = 0.0f, vh = 0.0f;
      if (node < NN && k < din) {
        va = agg[node * NSTRIDE + k];
        vh = hin[node * NSTRIDE + k];
      }
      sAgg[r * KPP + k] = va;
      sH[r * KPP + k]   = vh;
    }
  }
  __syncthreads();

  const int wave  = tid >> 5;
  const int lane  = tid & 31;
  const int nsub  = lane & 15;           // N index within 16-wide tile (B/C/D)
  const int khalf = (lane >> 4) << 1;    // 0 | 2 : K sub-offset (A/B frags)
  const int arow  = wave * 16 + nsub;    // A row within block (lane = M)
  const int mynode = row0 + arow;
  const float invd = (mynode < NN) ? 1.0f / fmaxf(deg[mynode], 1.0f) : 1.0f;
  const int npair = nsub * 2;

  // Accumulators, seeded with the bias (C/D: same N for all 8 sub-rows).
  v8f acc[CT];
#pragma unroll
  for (int ct = 0; ct < CT; ++ct) {
    int n = ct * 16 + nsub;
    float bv = (n < dout) ? bias[n] : 0.0f;
#pragma unroll
    for (int v = 0; v < 8; ++v) acc[ct][v] = bv;
  }

  // Panel 1: (agg * invd) x Wl   (Wp rows [0, KPP))
#pragma unroll 2
  for (int k0 = 0; k0 < KPP; k0 += 4) {
    v2f a = *(const v2f*)&sAgg[arow * KPP + k0 + khalf];
    a.x *= invd;
    a.y *= invd;
    const float* wrow = &Wp[((k0 + khalf) >> 1) * (2 * WCOLS) + npair];
    v2f b[CT];
#pragma unroll
    for (int ct = 0; ct < CT; ++ct) b[ct] = *(const v2f*)&wrow[ct * 32];
#pragma unroll
    for (int ct = 0; ct < CT; ++ct)
      acc[ct] = __builtin_amdgcn_wmma_f32_16x16x4_f32(
          false, a, false, b[ct], (short)0, acc[ct], false, false);
  }
  // Panel 2: hin x Wr   (Wp rows [KPP, 2*KPP))
  const float* Wp2 = Wp + (KPP >> 1) * (2 * WCOLS);
#pragma unroll 2
  for (int k0 = 0; k0 < KPP; k0 += 4) {
    v2f a = *(const v2f*)&sH[arow * KPP + k0 + khalf];
    const float* wrow = &Wp2[((k0 + khalf) >> 1) * (2 * WCOLS) + npair];
    v2f b[CT];
#pragma unroll
    for (int ct = 0; ct < CT; ++ct) b[ct] = *(const v2f*)&wrow[ct * 32];
#pragma unroll
    for (int ct = 0; ct < CT; ++ct)
      acc[ct] = __builtin_amdgcn_wmma_f32_16x16x4_f32(
          false, a, false, b[ct], (short)0, acc[ct], false, false);
  }

  // D layout: VGPR v = row (v | v+8) per lane half, col = lane&15.
  const int mbase = row0 + wave * 16 + ((lane >> 4) << 3);
#pragma unroll
  for (int ct = 0; ct < CT; ++ct) {
    int n = ct * 16 + nsub;
    if (n < dout) {
#pragma unroll
      for (int v = 0; v < 8; ++v) {
        int node = mbase + v;
        if (node < NN) hout[node * NSTRIDE + n] = acc[ct][v];
      }
    }
  }
}

// ---------------------------------------------------------------------------
// GraphNorm. batch ids are sorted, graphs have 1000 nodes -> a 256-node block
// spans at most 2 graphs: two-level reduction (ds_add_f32 then global atomic).
// ---------------------------------------------------------------------------
__global__ __launch_bounds__(256) void gnorm_mean_kernel(
    const float* __restrict__ h, const int* __restrict__ batch,
    float* __restrict__ gmean, int dout) {
  __shared__ float s[2 * GNPAD];
  const int n0 = blockIdx.x * 256;
  for (int i = threadIdx.x; i < 2 * GNPAD; i += 256) s[i] = 0.0f;
  __syncthreads();
  const int g0 = batch[n0 < NN ? n0 : NN - 1];
  const int node = n0 + threadIdx.x;
  if (node < NN) {
    int gg = batch[node] - g0;
    gg = gg < 0 ? 0 : (gg > 1 ? 1 : gg);
    for (int k = 0; k < dout; ++k)
      atomicAdd(&s[gg * GNPAD + k], h[node * NSTRIDE + k]);
  }
  __syncthreads();
  for (int i = threadIdx.x; i < 2 * GNPAD; i += 256) {
    int gg = i / GNPAD, k = i - gg * GNPAD;
    int g = g0 + gg;
    float v = s[i];
    if (k < dout && g < BB && v != 0.0f) atomicAdd(&gmean[g * MAXD + k], v);
  }
}

__global__ __launch_bounds__(256) void gnorm_center_kernel(
    float* __restrict__ h, const int* __restrict__ batch,
    const float* __restrict__ gmean, const float* __restrict__ gcnt,
    const float* __restrict__ alpha, float* __restrict__ gvar, int dout) {
  __shared__ float s[2 * GNPAD];
  const int n0 = blockIdx.x * 256;
  for (int i = threadIdx.x; i < 2 * GNPAD; i += 256) s[i] = 0.0f;
  __syncthreads();
  const int g0 = batch[n0 < NN ? n0 : NN - 1];
  const int node = n0 + threadIdx.x;
  if (node < NN) {
    int g = batch[node];
    int gg = g - g0;
    gg = gg < 0 ? 0 : (gg > 1 ? 1 : gg);
    float inv = 1.0f / gcnt[g];
    for (int k = 0; k < dout; ++k) {
      float c = h[node * NSTRIDE + k] - alpha[k] * gmean[g * MAXD + k] * inv;
      h[node * NSTRIDE + k] = c;
      atomicAdd(&s[gg * GNPAD + k], c * c);
    }
  }
  __syncthreads();
  for (int i = threadIdx.x; i < 2 * GNPAD; i += 256) {
    int gg = i / GNPAD, k = i - gg * GNPAD;
    int g = g0 + gg;
    float v = s[i];
    if (k < dout && g < BB && v != 0.0f) atomicAdd(&gvar[g * MAXD + k], v);
  }
}

__global__ __launch_bounds__(256) void gnorm_finish_kernel(
    float* __restrict__ h, const int* __restrict__ batch,
    const float* __restrict__ gvar, const float* __restrict__ gcnt,
    const float* __restrict__ scale, const float* __restrict__ shift, int dout) {
  int node = blockIdx.x * blockDim.x + threadIdx.x;
  if (node >= NN) return;
  int g = batch[node];
  float sn = floorf(sqrtf(gcnt[g]));    // int(sqrt(n))
  for (int k = 0; k < dout; ++k) {
    float r = sn * rsqrtf(gvar[g * MAXD + k]);   // c * sqrt_n / colnorm
    float v = h[node * NSTRIDE + k] * r * scale[k] + shift[k];
    h[node * NSTRIDE + k] = fmaxf(v, 0.0f);      // fused ReLU
  }
}

// ---------------------------------------------------------------------------
// Output conv (100 -> 1) + per-graph mean pooling
// ---------------------------------------------------------------------------
__global__ __launch_bounds__(256) void final_node_kernel(
    const float* __restrict__ h, const float* __restrict__ agg,
    const float* __restrict__ deg,
    const float* __restrict__ Wl, const float* __restrict__ bias,
    const float* __restrict__ Wr, const int* __restrict__ batch,
    float* __restrict__ gout) {
  int node = blockIdx.x * blockDim.x + threadIdx.x;
  if (node >= NN) return;
  float invd = 1.0f / fmaxf(deg[node], 1.0f);
  float acc = bias[0];
  const float4* ha = (const float4*)&h[node * NSTRIDE];
  const float4* aa = (const float4*)&agg[node * NSTRIDE];
  const float4* wl = (const float4*)Wl;
  const float4* wr = (const float4*)Wr;
  for (int c = 0; c < NSTRIDE / 4; ++c) {
    float4 hv = ha[c], av = aa[c], l = wl[c], r = wr[c];
    acc += (av.x * l.x + av.y * l.y + av.z * l.z + av.w * l.w) * invd;
    acc += hv.x * r.x + hv.y * r.y + hv.z * r.z + hv.w * r.w;
  }
  atomicAdd(&gout[batch[node]], acc);
}

__global__ __launch_bounds__(256) void final_div_kernel(
    const float* __restrict__ gout, const float* __restrict__ gcnt,
    float* __restrict__ out) {
  int g = blockIdx.x * blockDim.x + threadIdx.x;
  if (g < BB) out[g] = gout[g] / gcnt[g];
}

// ---------------------------------------------------------------------------
// Host launcher
// ---------------------------------------------------------------------------
static inline int cdiv(int a, int b) { return (a + b - 1) / b; }

extern "C" void kernel_launch(void* const* d_in, const int* in_sizes, int n_in,
                              void* d_out, int out_size, void* d_ws, size_t ws_size,
                              hipStream_t stream) {
  (void)in_sizes; (void)n_in; (void)out_size; (void)ws_size;

  const float* x     = (const float*)d_in[0];
  const int*   ei    = (const int*)d_in[1];
  const int*   batch = (const int*)d_in[2];
  const int*   src   = ei;           // edge_index[0]
  const int*   dst   = ei + EE;      // edge_index[1]

  const float *Wl[8], *Bs[8], *Wr[8];
  for (int i = 0; i < 7; ++i) {
    Wl[i] = (const float*)d_in[3 + 3 * i];
    Bs[i] = (const float*)d_in[4 + 3 * i];
    Wr[i] = (const float*)d_in[5 + 3 * i];
  }
  Wl[7] = (const float*)d_in[24];
  Bs[7] = (const float*)d_in[25];
  Wr[7] = (const float*)d_in[26];
  const float *alp[7], *scl[7], *shf[7];
  for (int i = 0; i < 7; ++i) {
    alp[i] = (const float*)d_in[27 + 3 * i];
    scl[i] = (const float*)d_in[28 + 3 * i];
    shf[i] = (const float*)d_in[29 + 3 * i];
  }

  // Workspace layout (floats): ~120.6 MB total, resident in the 192 MB L2.
  float* ws    = (float*)d_ws;
  float* hA    = ws;  ws += (size_t)NN * NSTRIDE;
  float* hB    = ws;  ws += (size_t)NN * NSTRIDE;
  float* agg   = ws;  ws += (size_t)NN * NSTRIDE;
  float* deg   = ws;  ws += NN;
  float* gcnt  = ws;  ws += BB;
  float* gmean = ws;  ws += BB * MAXD;
  float* gvar  = ws;  ws += BB * MAXD;
  float* gout  = ws;  ws += BB;
  float* Wp    = ws;  ws += 2 * MAXD * WCOLS;   // packed weight panel
  float* bufs[2] = { hA, hB };

  dim3 b256(256);

  // One-time per call: degrees, per-graph counts, input embedding.
  fill_kernel<<<cdiv(NN, 256), b256, 0, stream>>>(deg, 0.0f, NN);
  fill_kernel<<<cdiv(BB, 256), b256, 0, stream>>>(gcnt, 0.0f, BB);
  degree_kernel<<<cdiv(EE, 256), b256, 0, stream>>>(dst, deg);
  gcount_kernel<<<cdiv(NN, 256), b256, 0, stream>>>(batch, gcnt);
  init_h_kernel<<<cdiv(NN, 256), b256, 0, stream>>>(x, hA);

  static const int dims[8] = {1, 10, 100, 100, 100, 100, 100, 100};

  for (int i = 0; i < 7; ++i) {
    int din = dims[i], dout = dims[i + 1];
    int KPP = (din + 3) & ~3;
    float* hin  = bufs[i & 1];
    float* hout = bufs[(i + 1) & 1];

    fill_kernel<<<cdiv(NN * NSTRIDE, 256), b256, 0, stream>>>(agg, 0.0f, NN * NSTRIDE);
    if ((din & 3) == 0)
      aggregate_v4_kernel<<<dim3(cdiv(EE, 256), din / 4), b256, 0, stream>>>(hin, src, dst, agg);
    else
      aggregate_s_kernel<<<dim3(cdiv(EE, 256), din), b256, 0, stream>>>(hin, src, dst, agg);

    pack_w_kernel<<<cdiv(2 * KPP * WCOLS, 256), b256, 0, stream>>>(
        Wl[i], Wr[i], Wp, din, dout, KPP);
    if (dout <= 16)
      sage_gemm_wmma<1><<<cdiv(NN, TR), dim3(128), 0, stream>>>(
          hin, agg, deg, Wp, Bs[i], hout, din, dout, KPP);
    else
      sage_gemm_wmma<7><<<cdiv(NN, TR), dim3(128), 0, stream>>>(
          hin, agg, deg, Wp, Bs[i], hout, din, dout, KPP);

    fill_kernel<<<cdiv(BB * MAXD, 256), b256, 0, stream>>>(gmean, 0.0f, BB * MAXD);
    fill_kernel<<<cdiv(BB * MAXD, 256), b256, 0, stream>>>(gvar, 0.0f, BB * MAXD);
    gnorm_mean_kernel<<<cdiv(NN, 256), b256, 0, stream>>>(hout, batch, gmean, dout);
    gnorm_center_kernel<<<cdiv(NN, 256), b256, 0, stream>>>(hout, batch, gmean, gcnt,
                                                            alp[i], gvar, dout);
    gnorm_finish_kernel<<<cdiv(NN, 256), b256, 0, stream>>>(hout, batch, gvar, gcnt,
                                                            scl[i], shf[i], dout);
  }

  // Output conv (din=100, dout=1) + mean pool.
  float* hlast = bufs[7 & 1];
  fill_kernel<<<cdiv(NN * NSTRIDE, 256), b256, 0, stream>>>(agg, 0.0f, NN * NSTRIDE);
  aggregate_v4_kernel<<<dim3(cdiv(EE, 256), NSTRIDE / 4), b256, 0, stream>>>(hlast, src, dst, agg);
  fill_kernel<<<cdiv(BB, 256), b256, 0, stream>>>(gout, 0.0f, BB);
  final_node_kernel<<<cdiv(NN, 256), b256, 0, stream>>>(hlast, agg, deg,
                                                        Wl[7], Bs[7], Wr[7], batch, gout);
  final_div_kernel<<<cdiv(BB, 256), b256, 0, stream>>>(gout, gcnt, (float*)d_out);
}